// FPSampler_70007966925224
// MI455X (gfx1250) — compile-verified
//
#include <hip/hip_runtime.h>
#include <stdint.h>

#define N_POINTS   131072
#define N_SAMPLES  2048
#define NWG        32
#define NTHREADS   256
#define NWAVES     (NTHREADS / 32)
#define PPT        (N_POINTS / (NWG * NTHREADS))   // 16 points per thread

// d_ws layout (bytes):
//   0   : u64 key[2]   ping-pong packed (d2_bits<<32) | (~idx)
//   16  : u32 arrive   monotonic arrival counter
//   128 : u32 publish  monotonic published-iteration stamp
//   256 : u32 rec[4]   {winner idx, x, y, z}  (16B aligned)

__global__ void fps_init(unsigned char* ws, int* out) {
    if (blockIdx.x == 0 && threadIdx.x == 0) {
        ((unsigned long long*)ws)[0] = 0ull;
        ((unsigned long long*)ws)[1] = 0ull;
        *(unsigned int*)(ws + 16)  = 0u;
        *(unsigned int*)(ws + 128) = 0u;
        out[0] = 0;  // START_IDX
    }
}

__global__ __launch_bounds__(NTHREADS) void fps_main(const float* __restrict__ pos,
                                                     int* __restrict__ out,
                                                     unsigned char* __restrict__ ws) {
    unsigned long long* keys = (unsigned long long*)ws;
    unsigned int* arrive     = (unsigned int*)(ws + 16);
    unsigned int* publish    = (unsigned int*)(ws + 128);
    unsigned int* rec        = (unsigned int*)(ws + 256);

    __shared__ __align__(16) unsigned int bcast[4];
    __shared__ float swv[NWAVES];
    __shared__ int   swi[NWAVES];

    const int t    = threadIdx.x;
    const int wg   = blockIdx.x;
    const int gt   = wg * NTHREADS + t;
    const int base = gt * PPT;

    // ---- load this thread's slab of points into registers (float4 vectorized) ----
    float px[PPT], py[PPT], pz[PPT], d2[PPT];
    {
        const float4* src = reinterpret_cast<const float4*>(pos + (size_t)base * 3);
        float buf[PPT * 3];
        #pragma unroll
        for (int i = 0; i < (PPT * 3) / 4; ++i) {
            float4 v = src[i];
            buf[4*i+0] = v.x; buf[4*i+1] = v.y; buf[4*i+2] = v.z; buf[4*i+3] = v.w;
        }
        #pragma unroll
        for (int i = 0; i < PPT; ++i) {
            px[i] = buf[3*i]; py[i] = buf[3*i+1]; pz[i] = buf[3*i+2];
            d2[i] = 3.402823466e38f;   // +inf-ish; first update yields dist-to-point0
        }
    }

    float cx = pos[0], cy = pos[1], cz = pos[2];   // current (most recently selected) point

    for (int k = 0; k < N_SAMPLES - 1; ++k) {
        // ---- register-resident min-update + thread-local argmax (first occurrence) ----
        float bv = -1.0f; int bi = 0x7fffffff;
        #pragma unroll
        for (int i = 0; i < PPT; ++i) {
            float dx = px[i] - cx, dy = py[i] - cy, dz = pz[i] - cz;
            float nd = fmaf(dx, dx, fmaf(dy, dy, dz * dz));
            float v  = fminf(d2[i], nd);
            d2[i] = v;
            if (v > bv) { bv = v; bi = base + i; }
        }
        // ---- wave32 reduction (warpSize == 32 on gfx1250) ----
        #pragma unroll
        for (int off = 16; off > 0; off >>= 1) {
            float ov = __shfl_down(bv, off, 32);
            int   oi = __shfl_down(bi, off, 32);
            if (ov > bv || (ov == bv && oi < bi)) { bv = ov; bi = oi; }
        }
        // ---- cross-wave reduction via LDS ----
        if ((t & 31) == 0) { swv[t >> 5] = bv; swi[t >> 5] = bi; }
        __syncthreads();
        if (t == 0) {
            #pragma unroll
            for (int w = 1; w < NWAVES; ++w) {
                float ov = swv[w]; int oi = swi[w];
                if (ov > bv || (ov == bv && oi < bi)) { bv = ov; bi = oi; }
            }
            // pack: larger d2 wins; ties -> smaller index (d2 >= 0 so bits are monotone)
            unsigned long long keyv =
                ((unsigned long long)__float_as_uint(bv) << 32) |
                (unsigned long long)(0xffffffffu - (unsigned)bi);
            unsigned long long* slot = &keys[k & 1];
            __hip_atomic_fetch_max(slot, keyv, __ATOMIC_RELAXED, __HIP_MEMORY_SCOPE_AGENT);
            unsigned int prev = __hip_atomic_fetch_add(arrive, 1u, __ATOMIC_ACQ_REL,
                                                       __HIP_MEMORY_SCOPE_AGENT);
            if (prev == (unsigned)(NWG * (k + 1) - 1)) {
                // last arriver: decode winner, reset slot for iteration k+2, publish
                unsigned long long kk =
                    __hip_atomic_load(slot, __ATOMIC_RELAXED, __HIP_MEMORY_SCOPE_AGENT);
                __hip_atomic_store(slot, 0ull, __ATOMIC_RELAXED, __HIP_MEMORY_SCOPE_AGENT);
                int wi = (int)(0xffffffffu - (unsigned)(kk & 0xffffffffu));
                float wx = pos[3*wi+0], wy = pos[3*wi+1], wz = pos[3*wi+2];
                __hip_atomic_store(&rec[0], (unsigned)wi,          __ATOMIC_RELAXED, __HIP_MEMORY_SCOPE_AGENT);
                __hip_atomic_store(&rec[1], __float_as_uint(wx),   __ATOMIC_RELAXED, __HIP_MEMORY_SCOPE_AGENT);
                __hip_atomic_store(&rec[2], __float_as_uint(wy),   __ATOMIC_RELAXED, __HIP_MEMORY_SCOPE_AGENT);
                __hip_atomic_store(&rec[3], __float_as_uint(wz),   __ATOMIC_RELAXED, __HIP_MEMORY_SCOPE_AGENT);
                out[k + 1] = wi;
                __hip_atomic_store(publish, (unsigned)(k + 1), __ATOMIC_RELEASE,
                                   __HIP_MEMORY_SCOPE_AGENT);
            }
            // spin (with s_sleep) for publication of iteration k
            while (__hip_atomic_load(publish, __ATOMIC_ACQUIRE, __HIP_MEMORY_SCOPE_AGENT)
                   < (unsigned)(k + 1)) {
                __builtin_amdgcn_s_sleep(1);
            }
            // CDNA5 async path: DMA the 16B winner record straight into LDS,
            // tracked by ASYNCcnt (global_load_async_to_lds_b128 + s_wait_asynccnt).
            unsigned int       lds_off = (unsigned int)(uintptr_t)&bcast[0];
            unsigned long long gaddr   = (unsigned long long)(uintptr_t)rec;
            asm volatile("global_load_async_to_lds_b128 %0, %1, off scope:SCOPE_DEV\n\t"
                         "s_wait_asynccnt 0"
                         :: "v"(lds_off), "v"(gaddr) : "memory");
        }
        __syncthreads();
        cx = __uint_as_float(bcast[1]);
        cy = __uint_as_float(bcast[2]);
        cz = __uint_as_float(bcast[3]);
    }
}

extern "C" void kernel_launch(void* const* d_in, const int* in_sizes, int n_in,
                              void* d_out, int out_size, void* d_ws, size_t ws_size,
                              hipStream_t stream) {
    const float* pos = (const float*)d_in[0];   // [131072, 3] float32
    int* out         = (int*)d_out;             // [2048] int32 indices
    unsigned char* ws = (unsigned char*)d_ws;

    fps_init<<<1, 64, 0, stream>>>(ws, out);
    fps_main<<<NWG, NTHREADS, 0, stream>>>(pos, out, ws);
}